// MatrixMemory_37374805409862
// MI455X (gfx1250) — compile-verified
//
#include <hip/hip_runtime.h>

typedef __attribute__((ext_vector_type(2))) float v2f;
typedef __attribute__((ext_vector_type(8))) float v8f;

#define HEAD_DIM 128

// One workgroup (8 wave32) per batch element.
// Phase 1: vector updates + deterministic denom reduction.
// Phase 2: stream C -> C_new (rank-1 gated update) fused with out = C_new*q
//          via chained V_WMMA_F32_16X16X4_F32 (A = C_new tile, B = q broadcast).
__global__ __launch_bounds__(256) void mlstm_matmem_kernel(
    const float* __restrict__ k_in, const float* __restrict__ v_in,
    const float* __restrict__ q_in, const float* __restrict__ i_gate,
    const float* __restrict__ f_gate, const float* __restrict__ C_in,
    const float* __restrict__ n_in,
    float* __restrict__ out, float* __restrict__ C_new,
    float* __restrict__ n_new)
{
    __shared__ float ks_l[HEAD_DIM];   // k / sqrt(d)
    __shared__ float q_l[HEAD_DIM];
    __shared__ float v_l[HEAD_DIM];
    __shared__ float red[HEAD_DIM];
    __shared__ float denom_s;

    const int b = blockIdx.x;
    const int t = threadIdx.x;
    const float fg = f_gate[b];
    const float ig = i_gate[b];

    // ---- Phase 1: vectors, n_new, denominator ----
    if (t < HEAD_DIM) {
        const size_t off = (size_t)b * HEAD_DIM + t;
        const float kk = k_in[off] * 0.08838834764831845f;  // 1/sqrt(128)
        const float qq = q_in[off];
        const float nn = fg * n_in[off] + ig * kk;
        ks_l[t] = kk;
        q_l[t]  = qq;
        v_l[t]  = v_in[off];
        n_new[off] = nn;
        red[t] = nn * qq;
    }
    __syncthreads();
    #pragma unroll
    for (int s = HEAD_DIM / 2; s > 0; s >>= 1) {
        if (t < s) red[t] += red[t + s];
        __syncthreads();
    }
    if (t == 0) denom_s = fmaxf(fabsf(red[0]), 1.0f);
    __syncthreads();
    const float inv_denom = 1.0f / denom_s;

    // ---- Phase 2: fused C stream + WMMA matvec ----
    const int lane = t & 31;
    const int wave = t >> 5;
    const int i0   = wave << 4;          // 16-row stripe per wave
    const int m    = lane & 15;          // M index within tile
    const int koff = (lane >> 4) << 1;   // lanes 16..31 hold K=2,3

    const int r = i0 + m;                // this lane's C row
    const float v_r = v_l[r];
    const size_t rowoff = ((size_t)b * HEAD_DIM + r) * HEAD_DIM;
    const v2f* __restrict__ Crow  = (const v2f*)(C_in  + rowoff);
    v2f*       __restrict__ CnRow = (v2f*)      (C_new + rowoff);

    v8f acc = {0.f, 0.f, 0.f, 0.f, 0.f, 0.f, 0.f, 0.f};

    #pragma unroll 4
    for (int j0 = 0; j0 < HEAD_DIM; j0 += 4) {
        const int col = j0 + koff;
        // temporal load: each 128B line reused over 8 k-steps via L0
        const v2f cc = Crow[col >> 1];
        v2f a;
        a.x = fg * cc.x + ig * (v_r * ks_l[col]);
        a.y = fg * cc.y + ig * (v_r * ks_l[col + 1]);
        // C_new is never read again -> non-temporal store
        __builtin_nontemporal_store(a, &CnRow[col >> 1]);
        // B[k][n] = q[j0+k] for all n (broadcast across N)
        v2f bq;
        bq.x = q_l[col];
        bq.y = q_l[col + 1];
        acc = __builtin_amdgcn_wmma_f32_16x16x4_f32(
            /*neg_a=*/false, a, /*neg_b=*/false, bq,
            /*c_mod=*/(short)0, acc, /*reuse_a=*/false, /*reuse_b=*/false);
    }

    // D layout: lanes 0-15 VGPR d = out[i0+d]; lanes 16-31 VGPR d = out[i0+8+d]
    if (m == 0) {
        float* op = out + (size_t)b * HEAD_DIM + i0 + ((lane >> 4) << 3);
        #pragma unroll
        for (int d = 0; d < 8; ++d) op[d] = acc[d] * inv_denom;
    }
}

extern "C" void kernel_launch(void* const* d_in, const int* in_sizes, int n_in,
                              void* d_out, int out_size, void* d_ws, size_t ws_size,
                              hipStream_t stream) {
    // setup_inputs order: k, v, q, i_gate, f_gate, C, n
    const float* k  = (const float*)d_in[0];
    const float* v  = (const float*)d_in[1];
    const float* q  = (const float*)d_in[2];
    const float* ig = (const float*)d_in[3];
    const float* fg = (const float*)d_in[4];
    const float* C  = (const float*)d_in[5];
    const float* n  = (const float*)d_in[6];

    const int batch = in_sizes[3];  // i_gate has BATCH elements

    // d_out = [out (B,128) | C_new (B,128,128) | n_new (B,128)] flat f32
    float* outp  = (float*)d_out;
    float* C_new = outp + (size_t)batch * HEAD_DIM;
    float* n_new = C_new + (size_t)batch * HEAD_DIM * HEAD_DIM;

    mlstm_matmem_kernel<<<batch, 256, 0, stream>>>(k, v, q, ig, fg, C, n,
                                                   outp, C_new, n_new);
}